// WKVMemory_67250597920914
// MI455X (gfx1250) — compile-verified
//
#include <hip/hip_runtime.h>
#include <cstdint>

// Problem shape (matches reference: B=8, T=4096, D=1024, all fp32)
#define B_   8
#define T_   4096
#define D_   1024

#define TPB  64                    // 2 waves/block -> 128 blocks, spreads over many WGPs
#define TC   8                     // time rows per double-buffered chunk
#define NCH  (T_ / TC)             // 512 chunks

// LDS byte layout: [buf(2)][tensor(2: v,k)][wave(TPB/32)][row(TC) x 128B]
#define ROW_B   128                       // 32 channels * 4B
#define WAVE_B  (TC * ROW_B)              // 1024 B per (buf,tensor,wave)
#define TEN_B   ((TPB / 32) * WAVE_B)     // 2048 B per (buf,tensor)
#define BUF_B   (2 * TEN_B)               // 4096 B per buf
#define SMEM_F  (2 * BUF_B / 4)           // 2048 floats = 8 KB total

// CDNA5 async global -> LDS copy (ASYNCcnt-tracked DMA, 16B per lane).
#define ASYNC_B128(ldsaddr, gaddr)                                             \
  asm volatile("global_load_async_to_lds_b128 %0, %1, off"                    \
               :: "v"(ldsaddr), "v"(gaddr) : "memory")

__global__ __launch_bounds__(TPB) void wkv_scan_kernel(
    const float* __restrict__ values,
    const float* __restrict__ importances,
    const float* __restrict__ log_gain,
    const float* __restrict__ log_decay,
    float* __restrict__ out)
{
    __shared__ float smem[SMEM_F];

    const int tid = threadIdx.x;
    const int w   = tid >> 5;          // wave within block
    const int j   = tid & 31;          // lane within wave

    const int bpb  = D_ / TPB;                       // blocks per batch (16)
    const int b    = blockIdx.x / bpb;
    const int dblk = (blockIdx.x % bpb) * TPB;
    const int d    = dblk + tid;                     // this thread's channel

    // Per-channel constants (reference: decay = exp(-exp(ld)), gain = exp(lg)-1)
    const float dec  = expf(-expf(log_decay[d]));
    const float gain = expf(log_gain[d]) - 1.0f;

    // ---- async-copy lane mapping: one b128 op = 32 lanes x 16B = 4 rows x 128B ----
    const int rr   = j >> 3;            // sub-row 0..3 within one instruction
    const int colb = (j & 7) << 4;      // byte offset within a 128B row

    const size_t batch_off = (size_t)b * T_ * D_;    // elements
    const size_t g0 = (size_t)rr * (D_ * 4) + (size_t)(dblk + w * 32) * 4 + colb;
    const size_t g1 = g0 + (size_t)4 * (D_ * 4);     // rows 4..7 of the chunk

    uint64_t gv0 = (uint64_t)(uintptr_t)((const char*)(values      + batch_off) + g0);
    uint64_t gv1 = (uint64_t)(uintptr_t)((const char*)(values      + batch_off) + g1);
    uint64_t gk0 = (uint64_t)(uintptr_t)((const char*)(importances + batch_off) + g0);
    uint64_t gk1 = (uint64_t)(uintptr_t)((const char*)(importances + batch_off) + g1);
    const uint64_t gstep = (uint64_t)TC * D_ * 4;    // bytes between chunks

    // Low 32 bits of a flat shared pointer == workgroup-relative LDS byte offset.
    const uint32_t lds_base = (uint32_t)(uintptr_t)(&smem[0]);
    const uint32_t lane_l   = (uint32_t)(w * WAVE_B + rr * ROW_B + colb);

    // Prologue: stage chunk 0 into buffer 0 (4 async ops per wave).
    {
        uint32_t l = lds_base + lane_l;
        ASYNC_B128(l,                     gv0);
        ASYNC_B128(l + 4 * ROW_B,         gv1);
        ASYNC_B128(l + TEN_B,             gk0);
        ASYNC_B128(l + TEN_B + 4 * ROW_B, gk1);
    }

    float c = 0.0f, n = 0.0f;
    const int tread = w * (WAVE_B / 4) + j;          // float idx of row 0 for this thread
    float* orow = out + batch_off + d;

    for (int ci = 0; ci < NCH; ++ci) {
        if (ci + 1 < NCH) {
            // Issue next chunk into the other buffer, then wait for current chunk.
            gv0 += gstep; gv1 += gstep; gk0 += gstep; gk1 += gstep;
            uint32_t l = lds_base + (uint32_t)(((ci + 1) & 1) * BUF_B) + lane_l;
            ASYNC_B128(l,                     gv0);
            ASYNC_B128(l + 4 * ROW_B,         gv1);
            ASYNC_B128(l + TEN_B,             gk0);
            ASYNC_B128(l + TEN_B + 4 * ROW_B, gk1);
            // ASYNCcnt completes in order per wave: <=4 outstanding means the
            // 4 loads of the *current* chunk have landed in LDS.
            asm volatile("s_wait_asynccnt 4" ::: "memory");
        } else {
            asm volatile("s_wait_asynccnt 0" ::: "memory");
        }

        // Wave-private LDS regions: no cross-wave sharing -> no s_barrier needed.
        const float* sv = smem + (ci & 1) * (BUF_B / 4) + tread;
        const float* sk = sv + (TEN_B / 4);

        #pragma unroll
        for (int r = 0; r < TC; ++r) {
            float vv = sv[r * 32];
            float kk = sk[r * 32];
            float kv = kk * vv;
            c = fmaf(c, dec, kv);
            n = fmaf(n, dec, kk);
            float num = fmaf(gain, kv, c);
            float den = fmaf(gain, kk, n);   // strictly positive (k>0, n>=0, gain>=0)
            // NT store: don't let the 128MB output stream evict k/v from L2.
            __builtin_nontemporal_store(num * __builtin_amdgcn_rcpf(den),
                                        orow + (size_t)r * D_);
        }
        orow += (size_t)TC * D_;
    }
}

extern "C" void kernel_launch(void* const* d_in, const int* in_sizes, int n_in,
                              void* d_out, int out_size, void* d_ws, size_t ws_size,
                              hipStream_t stream) {
    const float* values      = (const float*)d_in[0];
    const float* importances = (const float*)d_in[1];
    const float* log_gain    = (const float*)d_in[2];
    const float* log_decay   = (const float*)d_in[3];
    float* out = (float*)d_out;

    dim3 grid(B_ * (D_ / TPB));   // 128 blocks
    dim3 block(TPB);              // 64 threads (2 waves)
    hipLaunchKernelGGL(wkv_scan_kernel, grid, block, 0, stream,
                       values, importances, log_gain, log_decay, out);
}